// NeuralKnotNet_35184372089093
// MI455X (gfx1250) — compile-verified
//
#include <hip/hip_runtime.h>
#include <math.h>

#define NN 50000
#define EE 800000
#define GG 256
#define CC 20
#define TT 4
#define OT 5                         // C / T
#define AVG_DEG_LOG 2.8332133440562162f  // log(17)
#define EPS_BN 1e-5f

typedef float v2f __attribute__((ext_vector_type(2)));
typedef float v8f __attribute__((ext_vector_type(8)));

// ---- monotone float <-> uint encoding so integer atomicMin/Max == float min/max ----
__device__ __forceinline__ unsigned ordf(float f) {
  unsigned u = __float_as_uint(f);
  return (u & 0x80000000u) ? ~u : (u | 0x80000000u);
}
__device__ __forceinline__ float unordf(unsigned u) {
  unsigned b = (u & 0x80000000u) ? (u ^ 0x80000000u) : ~u;
  return __uint_as_float(b);
}

// -------------------- degree / graph counts --------------------
__global__ void k_degree(const int* __restrict__ ei, float* deg, int E) {
  int e = blockIdx.x * blockDim.x + threadIdx.x;
  if (e < E) atomicAdd(&deg[ei[E + e]], 1.0f);   // dst = edge_index[1]
}
__global__ void k_gcnt(const int* __restrict__ batch, float* gcnt, int n) {
  int i = blockIdx.x * blockDim.x + threadIdx.x;
  if (i < n) atomicAdd(&gcnt[batch[i]], 1.0f);
}

// -------------------- layer 0 edge kernel (F = 1, trivial) --------------------
__global__ void k_edge_f1(const float* __restrict__ x, const int* __restrict__ ei,
                          const float* __restrict__ ea,
                          const float* __restrict__ eW, const float* __restrict__ eb,
                          const float* __restrict__ preW, const float* __restrict__ preb,
                          float* sum, float* sumsq, unsigned* mnb, unsigned* mxb, int E) {
  int e = blockIdx.x * blockDim.x + threadIdx.x;
  if (e >= E) return;
  int s = ei[e], d = ei[E + e];
  float h0 = x[d], h1 = x[s], h2 = ea[e] * eW[0] + eb[0];
#pragma unroll
  for (int t = 0; t < TT; ++t) {
    float m = preb[t] + h0 * preW[t * 3 + 0] + h1 * preW[t * 3 + 1] + h2 * preW[t * 3 + 2];
    int idx = d * 4 + t;                 // TF = 4 for layer 0
    atomicAdd(&sum[idx], m);
    atomicAdd(&sumsq[idx], m * m);
    unsigned o = ordf(m);
    atomicMax(&mxb[idx], o);
    atomicMin(&mnb[idx], o);
  }
}

// -------------------- layers 1-4 edge kernel: [E,60]x[60,80] via V_WMMA_F32_16X16X4_F32 ----
// block = 256 threads = 8 waves, 16 edges/wave -> 128 edges/block (E % 128 == 0)
__global__ void __launch_bounds__(256)
k_edge_wmma(const float* __restrict__ x, const int* __restrict__ ei,
            const float* __restrict__ ea,
            const float* __restrict__ eW, const float* __restrict__ eb,
            const float* __restrict__ preW, const float* __restrict__ preb,
            float* sum, float* sumsq, unsigned* mnb, unsigned* mxb, int E) {
  __shared__ float hT[128][68];   // h rows (K padded 60->64), stride 68 avoids bank conflicts
  __shared__ float WT[80][68];    // W^T : WT[n][k] = preW[t][k][o], n = t*20+o
  __shared__ int   sdst[128];
  __shared__ int   ssrc[128];
  __shared__ float sea[128];

  const int tid = threadIdx.x;
  const int ebase = blockIdx.x * 128;

  for (int i = tid; i < 80 * 64; i += 256) {
    int n = i / 64, k = i % 64;
    float w = 0.f;
    if (k < 60) { int t = n / 20, o = n % 20; w = preW[(t * 60 + k) * 20 + o]; }
    WT[n][k] = w;
  }
  if (tid < 128) {
    int e = ebase + tid; e = e < E ? e : (E - 1);
    ssrc[tid] = ei[e]; sdst[tid] = ei[E + e]; sea[tid] = ea[e];
  }
  __syncthreads();

  // stage h = [x_dst(20) | x_src(20) | edge_enc(20) | pad(4)]
  for (int i = tid; i < 128 * 64; i += 256) {
    int r = i >> 6, k = i & 63;
    float v = 0.f;
    if (k < 20)       v = x[sdst[r] * 20 + k];
    else if (k < 40)  v = x[ssrc[r] * 20 + (k - 20)];
    else if (k < 60)  { int f = k - 40; v = sea[r] * eW[f] + eb[f]; }
    hT[r][k] = v;
  }
  __syncthreads();

  const int wave = tid >> 5, lane = tid & 31, half = lane >> 4, l16 = lane & 15;
  const int wbase = wave * 16;

  v8f acc[5];
#pragma unroll
  for (int nt = 0; nt < 5; ++nt) {
    v8f z = {0.f, 0.f, 0.f, 0.f, 0.f, 0.f, 0.f, 0.f};
    acc[nt] = z;
  }

  // A 16x4 f32 layout: lanes 0-15 hold K={k0,k0+1}, lanes 16-31 hold K={k0+2,k0+3}, M=lane%16
  // B 4x16 f32 layout (mirrored): lane reads B[k][n] with same K split, N=lane%16
  for (int ks = 0; ks < 16; ++ks) {
    int kk = ks * 4 + 2 * half;
    v2f a = { hT[wbase + l16][kk], hT[wbase + l16][kk + 1] };
#pragma unroll
    for (int nt = 0; nt < 5; ++nt) {
      int col = nt * 16 + l16;
      v2f b = { WT[col][kk], WT[col][kk + 1] };
      acc[nt] = __builtin_amdgcn_wmma_f32_16x16x4_f32(
          false, a, false, b, (short)0, acc[nt], false, false);
    }
  }

  // D layout: vgpr j = row (j + 8*half), lane%16 = column -> add bias, scatter atomics
#pragma unroll
  for (int nt = 0; nt < 5; ++nt) {
    int c = nt * 16 + l16;             // output channel 0..79 (= t*20+f)
    float bias = preb[c];
#pragma unroll
    for (int j = 0; j < 8; ++j) {
      int r = j + 8 * half;
      int e = ebase + wbase + r;
      if (e < E) {
        float v = acc[nt][j] + bias;
        int node = sdst[wbase + r];
        int idx = node * 80 + c;
        atomicAdd(&sum[idx], v);
        atomicAdd(&sumsq[idx], v * v);
        unsigned ov = ordf(v);
        atomicMax(&mxb[idx], ov);
        atomicMin(&mnb[idx], ov);
      }
    }
  }
}

// -------------------- fused node kernel: aggr + scalers + post NN + lin + BN partials ----
__global__ void __launch_bounds__(256)
k_node(const float* __restrict__ xin, int F,
       const float* __restrict__ deg,
       const float* __restrict__ sum, const float* __restrict__ sumsq,
       const unsigned* __restrict__ mnb, const unsigned* __restrict__ mxb,
       const float* __restrict__ postW, const float* __restrict__ postb,
       const float* __restrict__ linW, const float* __restrict__ linb,
       float* __restrict__ ybuf, float* bnstat, int N) {
  const int NB = 16;
  __shared__ float aggrL[NB][TT * 4 * CC];  // up to 16 x 320
  __shared__ float zL[NB][CC];
  __shared__ float sAmp[NB], sAtt[NB], sDen[NB];
  __shared__ int   sHas[NB];
  __shared__ float psum[CC], psq[CC];

  const int tid = threadIdx.x;
  const int nb = blockIdx.x * NB;

  if (tid < NB) {
    int gn = nb + tid;
    float c = (gn < N) ? deg[gn] : 0.f;
    float dc = fmaxf(c, 1.f);
    float lg = logf(dc + 1.f);
    sDen[tid] = dc;
    sAmp[tid] = lg / AVG_DEG_LOG;
    sAtt[tid] = AVG_DEG_LOG / lg;
    sHas[tid] = (c > 0.f);
  }
  if (tid < CC) { psum[tid] = 0.f; psq[tid] = 0.f; }
  __syncthreads();

  const int F4 = 4 * F, AG = TT * F4, TF = TT * F;

  for (int w = tid; w < NB * AG; w += blockDim.x) {
    int nl = w / AG, j = w % AG;
    int t = j / F4, jj = j % F4, seg = jj / F, f = jj % F;
    int gn = nb + nl;
    float v = 0.f;
    if (gn < N) {
      int idx = gn * TF + t * F + f;
      float den = sDen[nl];
      if (seg == 0)      v = sum[idx] / den;
      else if (seg == 1) v = sHas[nl] ? unordf(mnb[idx]) : 0.f;
      else if (seg == 2) v = sHas[nl] ? unordf(mxb[idx]) : 0.f;
      else {
        float mean = sum[idx] / den, m2 = sumsq[idx] / den;
        float d2 = m2 - mean * mean;
        v = sqrtf(fmaxf(d2, 0.f) + 1e-5f);
      }
    }
    aggrL[nl][j] = v;
  }
  __syncthreads();

  const int F13 = 13 * F;
  for (int w = tid; w < NB * CC; w += blockDim.x) {
    int nl = w / CC, co = w % CC;
    int gn = nb + nl;
    int t = co / OT, o = co % OT;
    const float* Wt = postW + (size_t)(t * F13) * OT;
    float acc = postb[t * OT + o];
    if (gn < N) {
      float amp = sAmp[nl], att = sAtt[nl];
      for (int f = 0; f < F; ++f) acc += xin[gn * F + f] * Wt[f * OT + o];
      const float* A = aggrL[nl] + t * F4;
      for (int jj = 0; jj < F4; ++jj) {
        float a  = A[jj];
        float w1 = Wt[(F + jj) * OT + o];
        float w2 = Wt[(F + F4 + jj) * OT + o];
        float w3 = Wt[(F + 2 * F4 + jj) * OT + o];
        acc += a * (w1 + amp * w2 + att * w3);
      }
    }
    zL[nl][co] = acc;
  }
  __syncthreads();

  for (int w = tid; w < NB * CC; w += blockDim.x) {
    int nl = w / CC, co = w % CC;
    int gn = nb + nl;
    float y = linb[co];
    for (int k = 0; k < CC; ++k) y += zL[nl][k] * linW[k * CC + co];
    if (gn < N) {
      ybuf[gn * CC + co] = y;
      atomicAdd(&psum[co], y);
      atomicAdd(&psq[co], y * y);
    }
  }
  __syncthreads();
  if (tid < CC) {
    atomicAdd(&bnstat[tid], psum[tid]);
    atomicAdd(&bnstat[CC + tid], psq[tid]);
  }
}

// -------------------- BatchNorm finalize + apply --------------------
__global__ void k_bn_fin(const float* bnstat, const float* __restrict__ g,
                         const float* __restrict__ b, float* ab, int N) {
  int c = threadIdx.x;
  if (c < CC) {
    float mu  = bnstat[c] / (float)N;
    float var = bnstat[CC + c] / (float)N - mu * mu;
    float sc  = g[c] * rsqrtf(var + EPS_BN);
    ab[c] = sc; ab[CC + c] = b[c] - mu * sc;
  }
}
__global__ void k_bn_apply(const float* __restrict__ y, const float* __restrict__ ab,
                           float* __restrict__ xo, int n) {
  int i = blockIdx.x * blockDim.x + threadIdx.x;
  if (i < n) { int c = i % CC; xo[i] = tanhf(y[i] * ab[c] + ab[CC + c]); }
}

// -------------------- global pooling --------------------
__global__ void k_pool(const float* __restrict__ x, const int* __restrict__ batch,
                       float* gsum, unsigned* gmax, int n) {
  int i = blockIdx.x * blockDim.x + threadIdx.x;
  if (i < n) {
    int nd = i / CC, c = i % CC;
    float v = x[i];
    int g = batch[nd];
    atomicAdd(&gsum[g * CC + c], v);
    atomicMax(&gmax[g * CC + c], ordf(v));
  }
}

// -------------------- final MLP (one block, 256 graphs) --------------------
__global__ void __launch_bounds__(256)
k_mlp(const float* __restrict__ gsum, const unsigned* __restrict__ gmax,
      const float* __restrict__ gcnt,
      const float* W0, const float* b0, const float* bng, const float* bnb,
      const float* W1, const float* b1, const float* W2, const float* b2,
      const float* W3, const float* b3, const float* W4, const float* b4,
      float* __restrict__ out) {
  __shared__ float hb[GG][26];
  __shared__ float sc[25], sh[25];
  int g = threadIdx.x;

  float h[3 * CC];
  float cnt = gcnt[g];
  float den = fmaxf(cnt, 1.f);
  int has = (cnt > 0.f);
  for (int c = 0; c < CC; ++c) {
    float s = gsum[g * CC + c];
    h[c]          = has ? unordf(gmax[g * CC + c]) : 0.f;  // max
    h[CC + c]     = s / den;                                // mean
    h[2 * CC + c] = s;                                      // sum
  }
  for (int j = 0; j < 25; ++j) {
    float a = b0[j];
    for (int k = 0; k < 60; ++k) a += h[k] * W0[k * 25 + j];
    hb[g][j] = a;
  }
  __syncthreads();
  if (g < 25) {
    float s = 0.f, ss = 0.f;
    for (int r = 0; r < GG; ++r) { float v = hb[r][g]; s += v; ss += v * v; }
    float mu = s / (float)GG, var = ss / (float)GG - mu * mu;
    float scl = bng[g] * rsqrtf(var + EPS_BN);
    sc[g] = scl; sh[g] = bnb[g] - mu * scl;
  }
  __syncthreads();
  float a0[25], a1[25];
  for (int j = 0; j < 25; ++j) a0[j] = tanhf(hb[g][j] * sc[j] + sh[j]);
  for (int j = 0; j < 25; ++j) { float a = b1[j]; for (int k = 0; k < 25; ++k) a += a0[k] * W1[k * 25 + j]; a1[j] = tanhf(a); }
  for (int j = 0; j < 25; ++j) { float a = b2[j]; for (int k = 0; k < 25; ++k) a += a1[k] * W2[k * 25 + j]; a0[j] = tanhf(a); }
  for (int j = 0; j < 25; ++j) { float a = b3[j]; for (int k = 0; k < 25; ++k) a += a0[k] * W3[k * 25 + j]; a1[j] = fmaxf(a, 0.f); }
  float o = b4[0];
  for (int k = 0; k < 25; ++k) o += a1[k] * W4[k];
  out[g] = o;
}

// ==================== host launcher ====================
extern "C" void kernel_launch(void* const* d_in, const int* in_sizes, int n_in,
                              void* d_out, int out_size, void* d_ws, size_t ws_size,
                              hipStream_t stream) {
  // discriminate flattening order: sorted pytree keys vs dict insertion order
  bool sorted = (n_in > 1 && in_sizes[1] == EE);

  const int *batch, *eidx; const float *eattr, *x0;
  const float *ceW[5], *ceb[5], *cpreW[5], *cpreb[5], *cpostW[5], *cpostb[5],
              *clinW[5], *clinb[5], *cbng[5], *cbnb[5];
  const float *W0, *b0, *mbng, *mbnb, *W1, *b1, *W2, *b2, *W3, *b3, *W4, *b4;

  if (sorted) {
    // order: batch, edge_attr, edge_index, params{convs[ bn_b,bn_g,edge_W,edge_b,lin_W,lin_b,post_W,post_b,pre_W,pre_b ]...,
    //        mlp[ W0..W4, b0..b4, bn_b, bn_g ]}, x
    batch = (const int*)d_in[0]; eattr = (const float*)d_in[1]; eidx = (const int*)d_in[2];
    for (int l = 0; l < 5; ++l) {
      void* const* p = d_in + 3 + l * 10;
      cbnb[l]  = (const float*)p[0]; cbng[l]  = (const float*)p[1];
      ceW[l]   = (const float*)p[2]; ceb[l]   = (const float*)p[3];
      clinW[l] = (const float*)p[4]; clinb[l] = (const float*)p[5];
      cpostW[l]= (const float*)p[6]; cpostb[l]= (const float*)p[7];
      cpreW[l] = (const float*)p[8]; cpreb[l] = (const float*)p[9];
    }
    void* const* m = d_in + 53;
    W0 = (const float*)m[0]; W1 = (const float*)m[1]; W2 = (const float*)m[2];
    W3 = (const float*)m[3]; W4 = (const float*)m[4];
    b0 = (const float*)m[5]; b1 = (const float*)m[6]; b2 = (const float*)m[7];
    b3 = (const float*)m[8]; b4 = (const float*)m[9];
    mbnb = (const float*)m[10]; mbng = (const float*)m[11];
    x0 = (const float*)d_in[65];
  } else {
    // insertion order: x, edge_index, edge_attr, batch, convs[ edge_W..bn_b ], mlp[ W0,b0,bn_g,bn_b,W1,b1,... ]
    x0 = (const float*)d_in[0]; eidx = (const int*)d_in[1];
    eattr = (const float*)d_in[2]; batch = (const int*)d_in[3];
    for (int l = 0; l < 5; ++l) {
      void* const* p = d_in + 4 + l * 10;
      ceW[l]   = (const float*)p[0]; ceb[l]   = (const float*)p[1];
      cpreW[l] = (const float*)p[2]; cpreb[l] = (const float*)p[3];
      cpostW[l]= (const float*)p[4]; cpostb[l]= (const float*)p[5];
      clinW[l] = (const float*)p[6]; clinb[l] = (const float*)p[7];
      cbng[l]  = (const float*)p[8]; cbnb[l]  = (const float*)p[9];
    }
    void* const* m = d_in + 54;
    W0 = (const float*)m[0]; b0 = (const float*)m[1];
    mbng = (const float*)m[2]; mbnb = (const float*)m[3];
    W1 = (const float*)m[4]; b1 = (const float*)m[5];
    W2 = (const float*)m[6]; b2 = (const float*)m[7];
    W3 = (const float*)m[8]; b3 = (const float*)m[9];
    W4 = (const float*)m[10]; b4 = (const float*)m[11];
  }

  // workspace bump allocator (~76 MB total)
  char* ws = (char*)d_ws; size_t off = 0;
  auto alloc = [&](size_t bytes) -> void* {
    void* p = ws + off; off = (off + bytes + 255) & ~(size_t)255; return p;
  };
  float*    sum   = (float*)alloc((size_t)NN * 80 * 4);
  float*    sumsq = (float*)alloc((size_t)NN * 80 * 4);
  unsigned* mnb   = (unsigned*)alloc((size_t)NN * 80 * 4);
  unsigned* mxb   = (unsigned*)alloc((size_t)NN * 80 * 4);
  float*    deg   = (float*)alloc((size_t)NN * 4);
  float*    xA    = (float*)alloc((size_t)NN * CC * 4);
  float*    xB    = (float*)alloc((size_t)NN * CC * 4);
  float*    ybuf  = (float*)alloc((size_t)NN * CC * 4);
  float*    bnstat= (float*)alloc(2 * CC * 4);
  float*    bnab  = (float*)alloc(2 * CC * 4);
  float*    gsum  = (float*)alloc((size_t)GG * CC * 4);
  unsigned* gmax  = (unsigned*)alloc((size_t)GG * CC * 4);
  float*    gcnt  = (float*)alloc((size_t)GG * 4);
  (void)ws_size; (void)out_size;

  hipMemsetAsync(deg, 0, (size_t)NN * 4, stream);
  hipMemsetAsync(gcnt, 0, (size_t)GG * 4, stream);
  k_degree<<<(EE + 255) / 256, 256, 0, stream>>>(eidx, deg, EE);
  k_gcnt<<<(NN + 255) / 256, 256, 0, stream>>>(batch, gcnt, NN);

  const float* xcur = x0;   // layer 0 input, stride F=1
  float* nxt = xA;
  for (int l = 0; l < 5; ++l) {
    hipMemsetAsync(sum,   0,    (size_t)NN * 80 * 4, stream);
    hipMemsetAsync(sumsq, 0,    (size_t)NN * 80 * 4, stream);
    hipMemsetAsync(mnb,   0xFF, (size_t)NN * 80 * 4, stream);  // ordered +max
    hipMemsetAsync(mxb,   0,    (size_t)NN * 80 * 4, stream);  // ordered -max
    hipMemsetAsync(bnstat, 0, 2 * CC * 4, stream);

    if (l == 0)
      k_edge_f1<<<(EE + 255) / 256, 256, 0, stream>>>(
          xcur, eidx, eattr, ceW[0], ceb[0], cpreW[0], cpreb[0],
          sum, sumsq, mnb, mxb, EE);
    else
      k_edge_wmma<<<EE / 128, 256, 0, stream>>>(
          xcur, eidx, eattr, ceW[l], ceb[l], cpreW[l], cpreb[l],
          sum, sumsq, mnb, mxb, EE);

    int F = (l == 0) ? 1 : CC;
    k_node<<<(NN + 15) / 16, 256, 0, stream>>>(
        xcur, F, deg, sum, sumsq, mnb, mxb,
        cpostW[l], cpostb[l], clinW[l], clinb[l], ybuf, bnstat, NN);

    k_bn_fin<<<1, 32, 0, stream>>>(bnstat, cbng[l], cbnb[l], bnab, NN);
    k_bn_apply<<<(NN * CC + 255) / 256, 256, 0, stream>>>(ybuf, bnab, nxt, NN * CC);

    xcur = nxt; nxt = (nxt == xA) ? xB : xA;
  }

  hipMemsetAsync(gsum, 0, (size_t)GG * CC * 4, stream);
  hipMemsetAsync(gmax, 0, (size_t)GG * CC * 4, stream);
  k_pool<<<(NN * CC + 255) / 256, 256, 0, stream>>>(xcur, batch, gsum, gmax, NN * CC);
  k_mlp<<<1, GG, 0, stream>>>(gsum, gmax, gcnt,
                              W0, b0, mbng, mbnb, W1, b1, W2, b2, W3, b3, W4, b4,
                              (float*)d_out);
}